// PatchworkWrapper_46445776339672
// MI455X (gfx1250) — compile-verified
//
#include <hip/hip_runtime.h>
#include <hip/hip_bf16.h>
#include <math.h>

// ---------------------------------------------------------------------------
// DCT-II (2^24, ortho) -> band |X| segment sums -> 16 watermark bits.
// Four-step FFT: 2^24 = 4096 x 4096.  Row FFTs (4096-pt) are done fully in
// LDS as three radix-16 levels, each level expressed as 16x16 f32 matmuls on
// the CDNA5 matrix unit (V_WMMA_F32_16X16X4_F32, full fp32 precision).
// Row data is pulled into LDS by the Tensor Data Mover (TENSOR_LOAD_TO_LDS,
// TENSORcnt / s_wait_tensorcnt) when the toolchain exposes the builtin.
// Memory-bound design: 2 tiled transposes + 2 FFT passes + 1 reduction pass
// ~= 1.3 GB HBM traffic ~= 60us at 23.3 TB/s.
// Workspace layout (needs 4 * 64MB + 128B):
//   Are, Aim, Bre, Bim : 2^24 floats each ; bins : 32 floats
// ---------------------------------------------------------------------------

typedef __attribute__((ext_vector_type(2))) float v2f;
typedef __attribute__((ext_vector_type(8))) float v8f;
typedef __attribute__((ext_vector_type(4))) unsigned int u32x4;
typedef __attribute__((ext_vector_type(4))) int i32x4;
typedef __attribute__((ext_vector_type(8))) int i32x8;

#define NTOT   16777216   // 2^24
#define NSIDE  4096
#define SI     3145728    // 3000 Hz bin
#define KEND   7340032    // si + Ls (half-open)
#define LS     4194304
#define HALF   2097152
#define TWO_PI 6.28318530717958647692f

#if defined(__has_builtin)
#if __has_builtin(__builtin_amdgcn_tensor_load_to_lds) && \
    __has_builtin(__builtin_amdgcn_s_wait_tensorcnt)
#define USE_TDM 1
#endif
#endif

__device__ inline v8f wmma4(v2f a, v2f b, v8f c) {
  // D(16x16,f32) += A(16x4,f32) * B(4x16,f32); args:
  // (neg_a, A, neg_b, B, c_mod, C, reuse_a, reuse_b)
  return __builtin_amdgcn_wmma_f32_16x16x4_f32(false, a, false, b, (short)0, c,
                                               false, false);
}

#ifdef USE_TDM
// 1D row copy via TDM: 4096 x 4B contiguous elements, global -> LDS.
// D# group0: count=1 | lds_addr | global_addr[56:0] | type=2.
// D# group1: data_size=4B, tensor_dim0=tile_dim0=4096, dim0_stride=4096.
__device__ inline void tdm_row_load(unsigned lds_addr, const float* gptr) {
  unsigned long long ga = (unsigned long long)(size_t)(const void*)gptr;
  u32x4 g0;
  g0.x = 1u;                                                  // count = 1
  g0.y = lds_addr;                                            // LDS byte addr
  g0.z = (unsigned)ga;                                        // gaddr[31:0]
  g0.w = (unsigned)((ga >> 32) & 0x01FFFFFFu) | 0x80000000u;  // [56:32]|type=2
  i32x8 g1;
  g1[0] = 0x00020000;        // workgroup_mask=0, data_size=2 (4 bytes)
  g1[1] = (int)0x10000000;   // tensor_dim0 = 4096 (bits[79:48] low half)
  g1[2] = 0x00010000;        // tensor_dim0 hi=0, tensor_dim1 = 1
  g1[3] = (int)0x10000000;   // tensor_dim1 hi=0, tile_dim0 = 4096
  g1[4] = 0x00000001;        // tile_dim1 = 1, tile_dim2 = 0
  g1[5] = 0x00001000;        // tensor_dim0_stride = 4096
  g1[6] = 0;
  g1[7] = 0;
  i32x4 z4 = {0, 0, 0, 0};   // groups 2/3 unused (tile_dim3/4 = 0)
#if __has_include(<hip/amd_detail/amd_gfx1250_TDM.h>)
  i32x8 z8 = {0, 0, 0, 0, 0, 0, 0, 0};
  __builtin_amdgcn_tensor_load_to_lds(g0, g1, z4, z4, z8, 0);
#else
  __builtin_amdgcn_tensor_load_to_lds(g0, g1, z4, z4, 0);
#endif
}
#endif

// ---- 1) DCT-II input reorder: v[n]=x[2n], v[N-1-n]=x[2n+1]; imag = 0 -------
__global__ __launch_bounds__(256) void k_reorder(const float* __restrict__ x,
                                                 float* __restrict__ vre,
                                                 float* __restrict__ vim) {
  unsigned n = blockIdx.x * 256u + threadIdx.x;
  float v;
  if (n < (NTOT / 2)) v = x[2u * n];
  else                v = x[2u * (NTOT - 1u - n) + 1u];
  vre[n] = v;
  vim[n] = 0.0f;
}

// ---- 2) tiled 4096x4096 transpose (re & im together) -----------------------
__global__ __launch_bounds__(256) void k_transpose(const float* __restrict__ sre,
                                                   const float* __restrict__ sim,
                                                   float* __restrict__ dre,
                                                   float* __restrict__ dim_) {
  __shared__ float tr[32][33];
  __shared__ float ti[32][33];
  const int bx = blockIdx.x * 32, by = blockIdx.y * 32;
  const int tx = threadIdx.x, ty = threadIdx.y;   // block (32, 8)
#pragma unroll
  for (int j = 0; j < 32; j += 8) {
    size_t src = (size_t)(by + ty + j) * NSIDE + (bx + tx);
    tr[ty + j][tx] = sre[src];
    ti[ty + j][tx] = sim[src];
  }
  __syncthreads();
#pragma unroll
  for (int j = 0; j < 32; j += 8) {
    size_t dst = (size_t)(bx + ty + j) * NSIDE + (by + tx);
    dre[dst] = tr[tx][ty + j];
    dim_[dst] = ti[tx][ty + j];
  }
}

// ---- 3) 4096-point complex FFT per row, radix-16 x3 via f32 WMMA -----------
// Level A: G[k1][q]  = sum_p in[q+256p]    W16^{p k1}, * e^{-2pi i qk1/4096}
// Level B: H[kb][qb] = sum_p row[qb+16p]   W16^{p kb}, * e^{-2pi i qb kb/256}
// Level C: R[k2]     = sum_qb c[kb*16+qb]  W16^{qb k2} -> pos = k1+16kb+256k2
// If twiddle_top: multiply output col c of row r by e^{-2pi i r c / 2^24}.
__global__ __launch_bounds__(256) void k_fft4096(const float* __restrict__ gre,
                                                 const float* __restrict__ gim,
                                                 float* __restrict__ ore,
                                                 float* __restrict__ oim,
                                                 int twiddle_top) {
  extern __shared__ float smem[];
  float* b0re = smem;
  float* b0im = smem + 4096;
  float* b1re = smem + 8192;
  float* b1im = smem + 12288;

  const int tid  = threadIdx.x;
  const int lane = tid & 31;
  const int wave = tid >> 5;          // 8 waves
  const int hi   = lane >> 4;         // WMMA half-wave
  const int lo   = lane & 15;
  const long row = blockIdx.x;
  const size_t rb = (size_t)row * NSIDE;

#ifdef USE_TDM
  // Tensor Data Mover: DMA both 16KB row halves into LDS; wave 0 issues,
  // waits on TENSORcnt, and the workgroup barrier releases everyone.
  if (tid < 32) {
    unsigned lds0 = (unsigned)(size_t)(void*)b0re;
    tdm_row_load(lds0, gre + rb);
    tdm_row_load(lds0 + 4096u * (unsigned)sizeof(float), gim + rb);
    __builtin_amdgcn_s_wait_tensorcnt(0);
  }
#else
  for (int i = tid; i < NSIDE; i += 256) {
    b0re[i] = gre[rb + i];
    b0im[i] = gim[rb + i];
  }
#endif
  __syncthreads();

  // Constant DFT16 A-operands, ISA 16x4 f32 layout: elem(m=lo, k=vgpr+2*hi).
  // W16^{p m} = cos(th) + i sin(th), th = -2pi p m / 16.
  v2f ac[4], asn[4], aps[4];
#pragma unroll
  for (int s = 0; s < 4; ++s) {
    const int kk = 4 * s + 2 * hi;
    float t0 = (-TWO_PI / 16.0f) * (float)(lo * kk);
    float t1 = (-TWO_PI / 16.0f) * (float)(lo * (kk + 1));
    float c0, s0, c1, s1;
    __sincosf(t0, &s0, &c0);
    __sincosf(t1, &s1, &c1);
    ac[s].x = c0;  ac[s].y = c1;    // cos
    asn[s].x = s0; asn[s].y = s1;   // sin (negative angle baked in)
    aps[s].x = -s0; aps[s].y = -s1; // -sin
  }

  // ---------------- Level A : stride-256 butterflies ----------------
  for (int t = wave; t < 16; t += 8) {
    const int q0 = t * 16;
    v8f yre = {0.f, 0.f, 0.f, 0.f, 0.f, 0.f, 0.f, 0.f};
    v8f yim = {0.f, 0.f, 0.f, 0.f, 0.f, 0.f, 0.f, 0.f};
#pragma unroll
    for (int s = 0; s < 4; ++s) {
      const int kk = 4 * s + 2 * hi;
      const int base = q0 + lo;
      v2f br, bi;
      br.x = b0re[base + 256 * kk]; br.y = b0re[base + 256 * (kk + 1)];
      bi.x = b0im[base + 256 * kk]; bi.y = b0im[base + 256 * (kk + 1)];
      yre = wmma4(ac[s], br, yre);   // cos * re
      yre = wmma4(aps[s], bi, yre);  // -sin * im
      yim = wmma4(ac[s], bi, yim);   // cos * im
      yim = wmma4(asn[s], br, yim);  // sin * re
    }
#pragma unroll
    for (int v = 0; v < 8; ++v) {
      const int m = v + 8 * hi;
      const int q = q0 + lo;
      float ang = (-TWO_PI / 4096.0f) * (float)(q * m);
      float c, sn;
      __sincosf(ang, &sn, &c);
      float xr = yre[v], xi = yim[v];
      b1re[m * 256 + q] = xr * c - xi * sn;
      b1im[m * 256 + q] = xr * sn + xi * c;
    }
  }
  __syncthreads();

  // ---------------- Level B : per 256-row, stride-16 ----------------
  for (int t = wave; t < 16; t += 8) {
    const int rbase = t * 256;
    v8f yre = {0.f, 0.f, 0.f, 0.f, 0.f, 0.f, 0.f, 0.f};
    v8f yim = {0.f, 0.f, 0.f, 0.f, 0.f, 0.f, 0.f, 0.f};
#pragma unroll
    for (int s = 0; s < 4; ++s) {
      const int kk = 4 * s + 2 * hi;
      v2f br, bi;
      br.x = b1re[rbase + lo + 16 * kk]; br.y = b1re[rbase + lo + 16 * (kk + 1)];
      bi.x = b1im[rbase + lo + 16 * kk]; bi.y = b1im[rbase + lo + 16 * (kk + 1)];
      yre = wmma4(ac[s], br, yre);
      yre = wmma4(aps[s], bi, yre);
      yim = wmma4(ac[s], bi, yim);
      yim = wmma4(asn[s], br, yim);
    }
#pragma unroll
    for (int v = 0; v < 8; ++v) {
      const int m = v + 8 * hi;
      float ang = (-TWO_PI / 256.0f) * (float)(lo * m);
      float c, sn;
      __sincosf(ang, &sn, &c);
      float xr = yre[v], xi = yim[v];
      b0re[rbase + m * 16 + lo] = xr * c - xi * sn;
      b0im[rbase + m * 16 + lo] = xr * sn + xi * c;
    }
  }
  __syncthreads();

  // ---------------- Level C : contiguous 16-pt, no twiddle ----------
  for (int t = wave; t < 16; t += 8) {
    const int rbase = t * 256;
    v8f yre = {0.f, 0.f, 0.f, 0.f, 0.f, 0.f, 0.f, 0.f};
    v8f yim = {0.f, 0.f, 0.f, 0.f, 0.f, 0.f, 0.f, 0.f};
#pragma unroll
    for (int s = 0; s < 4; ++s) {
      const int kk = 4 * s + 2 * hi;
      v2f br, bi;
      br.x = b0re[rbase + lo * 16 + kk]; br.y = b0re[rbase + lo * 16 + kk + 1];
      bi.x = b0im[rbase + lo * 16 + kk]; bi.y = b0im[rbase + lo * 16 + kk + 1];
      yre = wmma4(ac[s], br, yre);
      yre = wmma4(aps[s], bi, yre);
      yim = wmma4(ac[s], bi, yim);
      yim = wmma4(asn[s], br, yim);
    }
#pragma unroll
    for (int v = 0; v < 8; ++v) {
      const int m = v + 8 * hi;                    // k2b
      b1re[t + 16 * lo + 256 * m] = yre[v];        // pos = k1 + 16*kb + 256*k2
      b1im[t + 16 * lo + 256 * m] = yim[v];
    }
  }
  __syncthreads();

  // ---------------- write back (+ optional top-level twiddle) ------
  for (int i = tid; i < NSIDE; i += 256) {
    float xr = b1re[i], xi = b1im[i];
    if (twiddle_top) {
      long tt = (row * (long)i) & (NTOT - 1);      // exact: < 2^24 fits float
      float ang = (-TWO_PI / (float)NTOT) * (float)tt;
      float c, sn;
      __sincosf(ang, &sn, &c);
      float nr = xr * c - xi * sn;
      float ni = xr * sn + xi * c;
      xr = nr;
      xi = ni;
    }
    ore[rb + i] = xr;
    oim[rb + i] = xi;
  }
}

// ---- 4) band reduction: |Re(e^{-i pi k/2N} V[k])| into 32 segment bins -----
__global__ __launch_bounds__(256) void k_reduce(const float* __restrict__ re,
                                                const float* __restrict__ im,
                                                float* __restrict__ bins) {
  __shared__ float lb[32];
  if (threadIdx.x < 32) lb[threadIdx.x] = 0.0f;
  __syncthreads();
  const size_t base = (size_t)blockIdx.x * 256u * 16u + threadIdx.x;
#pragma unroll
  for (int j = 0; j < 16; ++j) {
    size_t u = base + (size_t)j * 256u;            // linear scan of C matrix
    __builtin_prefetch(re + u + 2048, 0, 0);       // global_prefetch_b8
    __builtin_prefetch(im + u + 2048, 0, 0);
    long k1 = (long)(u >> 12);
    long k2 = (long)(u & 4095);
    long k = k1 + (k2 << 12);                      // V[k] = C[k%4096][k/4096]
    if (k >= SI && k < KEND) {
      long i = k - SI;
      int seg = (int)((i < HALF) ? (i >> 16) : ((LS - 1 - i) >> 16));
      float th = ((float)M_PI / 33554432.0f) * (float)k;  // pi*k/(2N)
      float c, s;
      __sincosf(th, &s, &c);
      float val = c * re[u] + s * im[u];           // Re(e^{-i th} V[k])
      atomicAdd(&lb[seg], fabsf(val));
    }
  }
  __syncthreads();
  if (threadIdx.x < 32) atomicAdd(&bins[threadIdx.x], lb[threadIdx.x]);
}

__global__ void k_zero(float* __restrict__ bins) {
  if (threadIdx.x < 32) bins[threadIdx.x] = 0.0f;
}

// ---- 5) bits: ((m1-m2)*rj >= 0) ? 0 : 1 ------------------------------------
__global__ void k_final(const float* __restrict__ bins, float* __restrict__ out) {
  int j = threadIdx.x;
  if (j < 16) {
    const float scale = (2.0f / sqrtf(2.0f * (float)NTOT)) / 65536.0f; // /segLen
    float m1 = bins[2 * j] * scale * 0.5f;
    float m2 = bins[2 * j + 1] * scale * 0.5f;
    float rj = 0.195f * __expf(-0.08f * (float)(j + 1));
    out[j] = ((m1 - m2) * rj >= 0.0f) ? 0.0f : 1.0f;
  }
}

extern "C" void kernel_launch(void* const* d_in, const int* in_sizes, int n_in,
                              void* d_out, int out_size, void* d_ws, size_t ws_size,
                              hipStream_t stream) {
  (void)in_sizes; (void)n_in; (void)out_size; (void)ws_size;
  const float* x = (const float*)d_in[0];
  float* out = (float*)d_out;

  float* Are = (float*)d_ws;            // needs 4 * 2^24 floats + 32 floats
  float* Aim = Are + (size_t)NTOT;
  float* Bre = Aim + (size_t)NTOT;
  float* Bim = Bre + (size_t)NTOT;
  float* bins = Bim + (size_t)NTOT;

  dim3 tgrid(NSIDE / 32, NSIDE / 32);
  dim3 tblk(32, 8);
  const size_t fft_lds = 4u * NSIDE * sizeof(float);  // 64 KB ping-pong

  k_reorder<<<NTOT / 256, 256, 0, stream>>>(x, Are, Aim);
  k_transpose<<<tgrid, tblk, 0, stream>>>(Are, Aim, Bre, Bim);   // T[n2][n1]
  k_fft4096<<<NSIDE, 256, fft_lds, stream>>>(Bre, Bim, Are, Aim, 1);
  k_transpose<<<tgrid, tblk, 0, stream>>>(Are, Aim, Bre, Bim);   // U[k1][n2]
  k_fft4096<<<NSIDE, 256, fft_lds, stream>>>(Bre, Bim, Are, Aim, 0);
  k_zero<<<1, 32, 0, stream>>>(bins);
  k_reduce<<<NTOT / (256 * 16), 256, 0, stream>>>(Are, Aim, bins);
  k_final<<<1, 32, 0, stream>>>(bins, out);
}